// Quantizer_49959059587220
// MI455X (gfx1250) — compile-verified
//
#include <hip/hip_runtime.h>
#include <hip/hip_bf16.h>

// Element-wise LUT quantizer:  out = scale * (nearest_lut(x/scale + zero) - zero)
//
// MI455X reasoning: 16 B HBM traffic/element, ~268 MB total -> ~11.5 us floor
// at 23.3 TB/s; pure bandwidth kernel (no matrix op -> WMMA inapplicable).
// Strategy: NT B128 streaming (256 MB stream > 192 MB L2, zero reuse),
// gfx1250 global_prefetch one paired stride (2 x stride) ahead, LUT in SGPRs (s_load_b512) with
// midpoints on the scalar f32 ALU, and a 15-deep v_cmp/v_cndmask select chain
// (~42 VALU/elem incl. IEEE div) that stays under the HBM roofline.
// This round: 2x manual unroll of the grid-stride loop so each wave keeps
// 6 B128 loads in flight instead of 3 (memory-level parallelism for 23.3 TB/s).

typedef float v4f __attribute__((ext_vector_type(4)));

namespace {
constexpr int kBlock = 256;        // 8 wave32 waves per workgroup
constexpr int kVecPerThread = 4;   // 4 x float4 = 16 elements per thread
}

// Nearest value in a sorted 16-entry LUT via boundary-midpoint chain.
// result = l[k], k = #{ i : q > (l[i]+l[i+1])/2 }  (strict '>' reproduces
// argmin's first-occurrence tie rule: exact midpoint -> lower index).
__device__ __forceinline__ float nearest_lut(float q, const float* __restrict__ l,
                                             const float* __restrict__ m) {
    float best = l[0];
#pragma unroll
    for (int i = 0; i < 15; ++i) {
        best = (q > m[i]) ? l[i + 1] : best;   // v_cmp + v_cndmask
    }
    return best;
}

__device__ __forceinline__ v4f quant4(v4f xx, v4f ss, v4f zz,
                                      const float* __restrict__ l,
                                      const float* __restrict__ m) {
    v4f rr;
#pragma unroll
    for (int c = 0; c < 4; ++c) {
        // IEEE division to match the reference's bucket decisions exactly.
        float q  = xx[c] / ss[c] + zz[c];
        float qv = nearest_lut(q, l, m);
        rr[c] = ss[c] * (qv - zz[c]);
    }
    return rr;
}

__global__ __launch_bounds__(kBlock)
void quant_lut_vec_kernel(const float* __restrict__ x,
                          const float* __restrict__ scale,
                          const float* __restrict__ zero,
                          const float* __restrict__ lut,
                          float* __restrict__ out,
                          int nvec) {
    // Tiny uniform-addressed LUT: scalarizes to s_load_b512; midpoints lower
    // to s_add_f32 (scalar float ALU). Amortized over 16 elements/thread.
    float l[16];
#pragma unroll
    for (int i = 0; i < 16; ++i) l[i] = lut[i];
    float m[15];
#pragma unroll
    for (int i = 0; i < 15; ++i) m[i] = 0.5f * (l[i] + l[i + 1]);

    const v4f* xv = (const v4f*)x;
    const v4f* sv = (const v4f*)scale;
    const v4f* zv = (const v4f*)zero;
    v4f*       ov = (v4f*)out;

    const int stride = gridDim.x * blockDim.x;            // in float4 units
    int idx = blockIdx.x * blockDim.x + threadIdx.x;

    // Paired iterations: issue 6 B128 loads before any compute (2x MLP).
    for (; idx + stride < nvec; idx += 2 * stride) {
        const int idx1 = idx + stride;

        // gfx1250 global_prefetch_b8 one paired-stride ahead (speculative:
        // OOB translations silently dropped per ISA).
        __builtin_prefetch(&xv[idx + 2 * stride], 0, 0);
        __builtin_prefetch(&sv[idx + 2 * stride], 0, 0);
        __builtin_prefetch(&zv[idx + 2 * stride], 0, 0);

        v4f xa = __builtin_nontemporal_load(&xv[idx]);
        v4f sa = __builtin_nontemporal_load(&sv[idx]);
        v4f za = __builtin_nontemporal_load(&zv[idx]);
        v4f xb = __builtin_nontemporal_load(&xv[idx1]);
        v4f sb = __builtin_nontemporal_load(&sv[idx1]);
        v4f zb = __builtin_nontemporal_load(&zv[idx1]);

        v4f ra = quant4(xa, sa, za, l, m);
        v4f rb = quant4(xb, sb, zb, l, m);

        __builtin_nontemporal_store(ra, &ov[idx]);
        __builtin_nontemporal_store(rb, &ov[idx1]);
    }
    // Remainder chunk (at most one per thread).
    for (; idx < nvec; idx += stride) {
        v4f xx = __builtin_nontemporal_load(&xv[idx]);
        v4f ss = __builtin_nontemporal_load(&sv[idx]);
        v4f zz = __builtin_nontemporal_load(&zv[idx]);
        v4f rr = quant4(xx, ss, zz, l, m);
        __builtin_nontemporal_store(rr, &ov[idx]);
    }
}

// Scalar tail for n % 4 != 0 (never runs for 4096x4096, kept for generality).
__global__ void quant_lut_tail_kernel(const float* __restrict__ x,
                                      const float* __restrict__ scale,
                                      const float* __restrict__ zero,
                                      const float* __restrict__ lut,
                                      float* __restrict__ out,
                                      int start, int n) {
    float l[16];
#pragma unroll
    for (int i = 0; i < 16; ++i) l[i] = lut[i];
    float m[15];
#pragma unroll
    for (int i = 0; i < 15; ++i) m[i] = 0.5f * (l[i] + l[i + 1]);

    int i = start + blockIdx.x * blockDim.x + threadIdx.x;
    if (i < n) {
        float s = scale[i], z = zero[i];
        float q = x[i] / s + z;
        out[i] = s * (nearest_lut(q, l, m) - z);
    }
}

extern "C" void kernel_launch(void* const* d_in, const int* in_sizes, int n_in,
                              void* d_out, int out_size, void* d_ws, size_t ws_size,
                              hipStream_t stream) {
    (void)n_in; (void)d_ws; (void)ws_size;
    const float* x     = (const float*)d_in[0];  // (N,D) f32
    const float* scale = (const float*)d_in[1];  // (N,D) f32
    const float* zero  = (const float*)d_in[2];  // (N,D) f32
    const float* lut   = (const float*)d_in[3];  // (16,) f32, sorted
    // d_in[4] = maxq (int scalar) — unused on the lookup path, as in reference.

    float* out = (float*)d_out;
    const long long n = (long long)out_size;     // == in_sizes[0]

    const int nvec = (int)(n >> 2);
    const int rem  = (int)(n & 3);

    if (nvec > 0) {
        const int vec_per_block = kBlock * kVecPerThread;               // 1024 float4s
        const int blocks = (nvec + vec_per_block - 1) / vec_per_block;  // 4096 for 16M elems
        quant_lut_vec_kernel<<<blocks, kBlock, 0, stream>>>(x, scale, zero, lut, out, nvec);
    }
    if (rem > 0) {
        quant_lut_tail_kernel<<<1, 64, 0, stream>>>(x, scale, zero, lut, out,
                                                    (int)(n - rem), (int)n);
    }
}